// MimicAcquisition_80917183856863
// MI455X (gfx1250) — compile-verified
//
#include <hip/hip_runtime.h>
#include <cstdint>

// MimicAcquisition, fused nearest(pass1) -> trilinear(pass2) resample.
// Per-axis index/weight math (all the f32 divides + RNE rounding) lives in a
// tiny LUT-building kernel (B*3*192 entries). The hot kernel stages the 4
// needed vol z-rows into LDS via gfx1250 async-to-LDS copies (ASYNCcnt path)
// and does the 8-corner weighted sum in the reference's exact fp order.

#define MAX_BATCH 64
// LUT entry per (batch, axis, index): {nmap(floor), nmap(ceil), w0, w1}
__device__ int4 g_lut[MAX_BATCH * 3 * 192];

__device__ __forceinline__ int nmap(int a, float dz) {
  // pass 1: down_loc = clip(a / dz, 0, 192); idx = clip(rint(down_loc), 0, 191)
  float dl = (float)a / dz;
  dl = fminf(dl, 192.0f);          // dl >= 0 always
  int n = (int)rintf(dl);          // RNE == jnp.round
  return n > 191 ? 191 : n;
}

// ---- setup: build per-(batch,axis,index) LUT ----
__global__ __launch_bounds__(192) void mimic_setup(const float* __restrict__ ssr) {
  const int t = threadIdx.x;            // output index along the axis
  const int bd = blockIdx.x;            // b*3 + d
  const int b = bd / 3, d = bd % 3;

  const float ss = ssr[b * 3 + d];
  const int dsi = (int)(192.0f / ss);   // int32 truncation as in reference
  const float f = (float)dsi;
  const float dz = f / 192.0f;          // down_zoom
  const float uz = 192.0f / f;          // up_zoom

  float loc = (float)t / uz;            // pass-2 sample location
  loc = fminf(fmaxf(loc, 0.0f), 191.0f);
  const float f0 = floorf(loc);
  const float f1 = fminf(f0 + 1.0f, 191.0f);
  const float w0 = f1 - loc;            // floor-corner weight
  const float w1 = 1.0f - w0;           // ceil-corner weight

  int4 e;
  e.x = nmap((int)f0, dz);              // composed pass-1 nearest index
  e.y = nmap((int)f1, dz);
  e.z = __float_as_int(w0);
  e.w = __float_as_int(w1);
  g_lut[bd * 192 + t] = e;
}

// ---- hot kernel: one block per (b, x, y) output row, 192 threads over z ----
__global__ __launch_bounds__(192) void mimic_main(
    const float* __restrict__ vol, float* __restrict__ out) {
  const int k = threadIdx.x;   // output z
  const int j = blockIdx.x;    // output y
  const int i = blockIdx.y;    // output x
  const int b = blockIdx.z;    // batch

  const int4 ex = g_lut[(b * 3 + 0) * 192 + i];   // block-uniform -> scalar
  const int4 ey = g_lut[(b * 3 + 1) * 192 + j];   // block-uniform -> scalar
  const int4 ez = g_lut[(b * 3 + 2) * 192 + k];   // per-thread

  const int nx0 = ex.x, nx1 = ex.y, ny0 = ey.x, ny1 = ey.y;
  const int nz0 = ez.x, nz1 = ez.y;
  const float wx0 = __int_as_float(ex.z), wx1 = __int_as_float(ex.w);
  const float wy0 = __int_as_float(ey.z), wy1 = __int_as_float(ey.w);
  const float wz0 = __int_as_float(ez.z), wz1 = __int_as_float(ez.w);

  // stage the 4 needed vol z-rows into LDS with async copies (ASYNCcnt path)
  __shared__ float rows[4][192];       // r = bx*2 + by
  const float* base = vol + (size_t)b * (192u * 192u * 192u);  // SGPR pair
  // uniform per-row byte offsets inside this batch's volume
  const uint32_t roff[4] = {
      (uint32_t)(nx0 * 192 + ny0) * 192u * 4u,
      (uint32_t)(nx0 * 192 + ny1) * 192u * 4u,
      (uint32_t)(nx1 * 192 + ny0) * 192u * 4u,
      (uint32_t)(nx1 * 192 + ny1) * 192u * 4u};
  const uint32_t k4 = (uint32_t)k * 4u;
  const uint32_t lds0 = (uint32_t)(uintptr_t)&rows[0][k];  // low 32 = LDS offset
#pragma unroll
  for (int r = 0; r < 4; ++r) {
    const uint32_t l = lds0 + (uint32_t)r * 768u;
    const uint32_t voff = roff[r] + k4;
    asm volatile("global_load_async_to_lds_b32 %0, %1, %2"
                 :: "v"(l), "v"(voff), "s"(base) : "memory");
  }
#if __has_builtin(__builtin_amdgcn_s_wait_asynccnt)
  __builtin_amdgcn_s_wait_asynccnt(0);
#else
  asm volatile("s_wait_asynccnt 0x0" ::: "memory");
#endif
  __syncthreads();

  // 8-corner accumulation in the reference's exact order (x bit fastest)
  const float wX[2] = {wx0, wx1}, wY[2] = {wy0, wy1}, wZ[2] = {wz0, wz1};
  const int lz[2] = {nz0, nz1};
  float acc = 0.0f;
#pragma unroll
  for (int c = 0; c < 8; ++c) {
    const int bx = c & 1, by = (c >> 1) & 1, bz = (c >> 2) & 1;
    const float w = (wX[bx] * wY[by]) * wZ[bz];
    acc += w * rows[bx * 2 + by][lz[bz]];
  }
  out[(((((size_t)b * 192 + i) * 192) + j) * 192 + k)] = acc;
}

extern "C" void kernel_launch(void* const* d_in, const int* in_sizes, int n_in,
                              void* d_out, int out_size, void* d_ws, size_t ws_size,
                              hipStream_t stream) {
  const float* vol = (const float*)d_in[0];   // [B,192,192,192,1] f32
  const float* ssr = (const float*)d_in[1];   // [B,3] f32
  float* out = (float*)d_out;                 // [B,192,192,192,1] f32
  int batch = in_sizes[1] / 3;
  if (batch > MAX_BATCH) batch = MAX_BATCH;   // reference uses BATCH=2
  mimic_setup<<<batch * 3, 192, 0, stream>>>(ssr);
  dim3 grid(192, 192, batch);
  mimic_main<<<grid, 192, 0, stream>>>(vol, out);
}